// LGCN_28819230556558
// MI455X (gfx1250) — compile-verified
//
#include <hip/hip_runtime.h>
#include <hip/hip_bf16.h>

// Problem constants (match reference)
#define NN   50000      // nodes
#define NR   47         // relations
#define NRP  52         // relations padded (covers rb+3 <= r1+2 <= 49; zero weights beyond NR)
#define EMB  1600       // embedding dim
#define KK   (EMB / 4)  // 400 WMMA K-steps
#define WS   16         // hidden width
#define NC   50         // classes
#define NB   40         // bases
#define NE   1600000    // edges

typedef __attribute__((ext_vector_type(2))) float v2f;
typedef __attribute__((ext_vector_type(8))) float v8f;

// ---------------- utility kernels ----------------

__global__ void k_zero(float* __restrict__ p, int n) {
    int i = blockIdx.x * blockDim.x + threadIdx.x;
    int st = gridDim.x * blockDim.x;
    for (; i < n; i += st) p[i] = 0.0f;
}

// cnt[c] = number of edges with cols[e] == c
__global__ void k_count(const int* __restrict__ cols, float* __restrict__ cnt) {
    int i = blockIdx.x * blockDim.x + threadIdx.x;
    int st = gridDim.x * blockDim.x;
    for (; i < NE; i += st) atomicAdd(&cnt[cols[i]], 1.0f);
}

// Generic basis combine (layer-2 weights): w[r, io] = sum_b comps[r,b]*bases[b,io]
__global__ void k_make_w(const float* __restrict__ comps, const float* __restrict__ bases,
                         float* __restrict__ w, int IO) {
    int i = blockIdx.x * blockDim.x + threadIdx.x;
    int st = gridDim.x * blockDim.x;
    int total = NR * IO;
    for (; i < total; i += st) {
        int r = i / IO, io = i - r * IO;
        float s = 0.0f;
        for (int b = 0; b < NB; ++b)
            s += comps[r * NB + b] * bases[(long)b * IO + io];
        w[i] = s;
    }
}

// Layer-1 weights pre-swizzled into the WMMA B-fragment layout (4x16 f32, ISA 7.12.2):
//   lane l<16  -> (K=4kk+0, o=l), (K=4kk+1, o=l)
//   lane l>=16 -> (K=4kk+2, o=l&15), (K=4kk+3, o=l&15)
// w1f[((r*KK + kk)*32 + lane)] : v2f, zero-padded for r >= NR.
__global__ void k_make_w1f(const float* __restrict__ comps, const float* __restrict__ bases,
                           v2f* __restrict__ w1f) {
    int i = blockIdx.x * blockDim.x + threadIdx.x;
    int st = gridDim.x * blockDim.x;
    const int total = NRP * KK * 32;
    for (; i < total; i += st) {
        int lane = i & 31;
        int kk   = (i >> 5) % KK;
        int r    = i / (KK * 32);
        int o = lane & 15, hi = lane >> 4;
        int k = 4 * kk + 2 * hi;
        float s0 = 0.0f, s1 = 0.0f;
        if (r < NR) {
            for (int b = 0; b < NB; ++b) {
                float c = comps[r * NB + b];
                const float* bp = bases + ((long)b * EMB + k) * WS + o;
                s0 += c * bp[0];
                s1 += c * bp[WS];
            }
        }
        v2f v; v.x = s0; v.y = s1;
        w1f[i] = v;
    }
}

// ---------------- layer-1 GEMM: H1[rr, n, o] = (1/cnt) * X[n,:] @ w1[r0+rr,:,o] -----
// One wave per (32-node tile pair, 4-relation group). Software-pipelined:
// prefetch next k-step's 6 fragments while 8 v_wmma_f32_16x16x4_f32 consume current.

__global__ __launch_bounds__(32)
void k_gemm1(const float* __restrict__ X, const v2f* __restrict__ w1f,
             const float* __restrict__ cnt, float* __restrict__ H1,
             int r0, int r1) {
    const int lane = threadIdx.x;
    const int col  = lane & 15;          // A: row m; B/C: column o
    const int hi   = lane >> 4;          // K half selector
    long n0 = (long)blockIdx.x * 32;     // first 16-node tile
    long n1 = n0 + 16;                   // second tile; duplicate n0 on tail (idempotent)
    if (n1 + 16 > NN) n1 = n0;
    const int rb = r0 + (int)blockIdx.y * 4;   // rb+3 < NRP always (padded weights)

    const float* xa0 = X + (n0 + col) * EMB + 2 * hi;
    const float* xa1 = X + (n1 + col) * EMB + 2 * hi;
    const v2f* bp0 = w1f + ((long)(rb + 0) * KK) * 32 + lane;
    const v2f* bp1 = w1f + ((long)(rb + 1) * KK) * 32 + lane;
    const v2f* bp2 = w1f + ((long)(rb + 2) * KK) * 32 + lane;
    const v2f* bp3 = w1f + ((long)(rb + 3) * KK) * 32 + lane;

    v8f c00 = {}, c01 = {}, c02 = {}, c03 = {};
    v8f c10 = {}, c11 = {}, c12 = {}, c13 = {};

#define WMMA(A, B, C) __builtin_amdgcn_wmma_f32_16x16x4_f32(false, A, false, B, (short)0, C, false, false)
#define WMMA8()                        \
    c00 = WMMA(a0, b0, c00);           \
    c10 = WMMA(a1, b0, c10);           \
    c01 = WMMA(a0, b1, c01);           \
    c11 = WMMA(a1, b1, c11);           \
    c02 = WMMA(a0, b2, c02);           \
    c12 = WMMA(a1, b2, c12);           \
    c03 = WMMA(a0, b3, c03);           \
    c13 = WMMA(a1, b3, c13);

    // prologue: fragments for kk = 0
    v2f a0 = *(const v2f*)(xa0);
    v2f a1 = *(const v2f*)(xa1);
    v2f b0 = bp0[0], b1 = bp1[0], b2 = bp2[0], b3 = bp3[0];

#pragma unroll 4
    for (int kk = 0; kk < KK - 1; ++kk) {
        // prefetch kk+1 fragments (independent of the WMMAs below)
        v2f na0 = *(const v2f*)(xa0 + 4 * (kk + 1));
        v2f na1 = *(const v2f*)(xa1 + 4 * (kk + 1));
        v2f nb0 = bp0[(kk + 1) * 32];
        v2f nb1 = bp1[(kk + 1) * 32];
        v2f nb2 = bp2[(kk + 1) * 32];
        v2f nb3 = bp3[(kk + 1) * 32];
        WMMA8()
        a0 = na0; a1 = na1;
        b0 = nb0; b1 = nb1; b2 = nb2; b3 = nb3;
    }
    WMMA8()   // peeled final k-step
#undef WMMA8
#undef WMMA

    // Epilogue: scale each output row by 1/cnt[r*NN + n] (== vals of the SpMM) and store.
    // C/D layout: VGPR v -> row v (lanes 0-15) / row v+8 (lanes 16-31); lane&15 -> col.
#define ST_REL(J, C0, C1)                                                           \
    if (rb + (J) < r1) {                                                            \
        const float* cA = cnt + (long)(rb + (J)) * NN + n0 + hi * 8;                \
        const float* cB = cnt + (long)(rb + (J)) * NN + n1 + hi * 8;                \
        float* opA = H1 + ((long)(rb - r0 + (J)) * NN + n0) * WS + col;             \
        float* opB = H1 + ((long)(rb - r0 + (J)) * NN + n1) * WS + col;             \
        _Pragma("unroll")                                                           \
        for (int v = 0; v < 8; ++v) {                                               \
            opA[(long)(v + hi * 8) * WS] = C0[v] / cA[v];                           \
            opB[(long)(v + hi * 8) * WS] = C1[v] / cB[v];                           \
        }                                                                           \
    }
    ST_REL(0, c00, c10)
    ST_REL(1, c01, c11)
    ST_REL(2, c02, c12)
    ST_REL(3, c03, c13)
#undef ST_REL
}

// ---------------- layer-1 scatter: h_mid[row] += H1[c]   (H1 pre-normalized) -------
__global__ void k_scatter1(const int* __restrict__ rows, const int* __restrict__ cols,
                           const float* __restrict__ H1, float* __restrict__ hmid,
                           int r0, int r1) {
    int i = blockIdx.x * blockDim.x + threadIdx.x;
    int st = gridDim.x * blockDim.x;
    const int total = NE * WS;                     // 25.6M
    for (; i < total; i += st) {
        int e = i >> 4, o = i & 15;
        int c = cols[e];
        int r = c / NN;
        if (r < r0 || r >= r1) continue;
        int cn = c - r * NN;
        float val = H1[((long)(r - r0) * NN + cn) * WS + o];
        atomicAdd(&hmid[(long)rows[e] * WS + o], val);
    }
}

__global__ void k_bias_relu(float* __restrict__ h, const float* __restrict__ bias1) {
    int i = blockIdx.x * blockDim.x + threadIdx.x;
    int st = gridDim.x * blockDim.x;
    const int total = NN * WS;
    for (; i < total; i += st)
        h[i] = fmaxf(h[i] + bias1[i & 15], 0.0f);
}

__global__ void k_init_out(float* __restrict__ out, const float* __restrict__ bias2) {
    int i = blockIdx.x * blockDim.x + threadIdx.x;
    int st = gridDim.x * blockDim.x;
    const int total = NN * NC;
    for (; i < total; i += st)
        out[i] = bias2[i % NC];
}

// ---------------- fused layer-2: per-edge (1x16)@(16x50), scaled scatter ----------------
__global__ void k_scatter2(const int* __restrict__ rows, const int* __restrict__ cols,
                           const float* __restrict__ cnt, const float* __restrict__ hmid,
                           const float* __restrict__ w2, float* __restrict__ out) {
    int i = blockIdx.x * blockDim.x + threadIdx.x;
    int st = gridDim.x * blockDim.x;
    const int total = NE * 64;                     // o padded to 64
    for (; i < total; i += st) {
        int e = i >> 6, o = i & 63;
        if (o >= NC) continue;
        int c = cols[e];
        int r = c / NN;
        int cn = c - r * NN;
        float v = 1.0f / cnt[c];
        const float* hp = hmid + (long)cn * WS;
        const float* wp = w2 + (long)r * (WS * NC) + o;
        float s = 0.0f;
#pragma unroll
        for (int t = 0; t < WS; ++t) s += hp[t] * wp[t * NC];
        atomicAdd(&out[(long)rows[e] * NC + o], v * s);
    }
}

// ---------------- host ----------------

static inline int gsz(long n) { return (int)((n + 255) / 256); }

extern "C" void kernel_launch(void* const* d_in, const int* in_sizes, int n_in,
                              void* d_out, int out_size, void* d_ws, size_t ws_size,
                              hipStream_t stream) {
    (void)in_sizes; (void)n_in; (void)out_size;
    const float* X      = (const float*)d_in[0];
    const float* comps1 = (const float*)d_in[1];
    const float* bases1 = (const float*)d_in[2];
    const float* comps2 = (const float*)d_in[3];
    const float* bases2 = (const float*)d_in[4];
    const float* bias1  = (const float*)d_in[5];
    const float* bias2  = (const float*)d_in[6];
    const int*   rows   = (const int*)d_in[7];
    const int*   cols   = (const int*)d_in[8];
    float*       out    = (float*)d_out;
    float*       ws     = (float*)d_ws;

    // Workspace layout
    size_t off = 0;
    float* cnt   = ws + off; off += (size_t)NR * NN;            // 2.35M f
    float* w1f   = ws + off; off += (size_t)NRP * KK * 32 * 2;  // 1.33M f (fragment-swizzled, padded)
    float* w2    = ws + off; off += (size_t)NR * WS * NC;       // 37.6K f
    float* hmid  = ws + off; off += (size_t)NN * WS;            // 0.80M f
    float* H1    = ws + off;                                    // Rc * NN * WS floats

    long remF = (long)(ws_size / sizeof(float)) - (long)off;
    long perRel = (long)NN * WS;
    int Rc = (int)(remF / perRel);
    if (Rc < 1)  Rc = 1;
    if (Rc > NR) Rc = NR;

    // normalization counts
    k_zero <<<gsz((long)NR * NN), 256, 0, stream>>>(cnt, NR * NN);
    k_count<<<gsz(NE),            256, 0, stream>>>(cols, cnt);

    // basis-combined weights
    k_make_w1f<<<gsz((long)NRP * KK * 32), 256, 0, stream>>>(comps1, bases1, (v2f*)w1f);
    k_make_w  <<<gsz((long)NR * WS * NC),  256, 0, stream>>>(comps2, bases2, w2, WS * NC);

    // layer 1: chunked WMMA GEMM (pre-normalized) + scatter-accumulate
    k_zero<<<gsz((long)NN * WS), 256, 0, stream>>>(hmid, NN * WS);
    const int nTilePairs = (NN / 16 + 1) / 2;   // 1563
    for (int rr0 = 0; rr0 < NR; rr0 += Rc) {
        int rr1 = rr0 + Rc; if (rr1 > NR) rr1 = NR;
        dim3 grid((unsigned)nTilePairs, (unsigned)((rr1 - rr0 + 3) / 4));
        k_gemm1<<<grid, 32, 0, stream>>>(X, (const v2f*)w1f, cnt, H1, rr0, rr1);
        k_scatter1<<<gsz((long)NE * WS), 256, 0, stream>>>(rows, cols, H1, hmid, rr0, rr1);
    }
    k_bias_relu<<<gsz((long)NN * WS), 256, 0, stream>>>(hmid, bias1);

    // layer 2: fused transform + scatter, output pre-seeded with bias2
    k_init_out<<<gsz((long)NN * NC), 256, 0, stream>>>(out, bias2);
    k_scatter2<<<gsz((long)NE * 64), 256, 0, stream>>>(rows, cols, cnt, hmid, w2, out);
}